// GraphSAGESudokuSolver_90580860273246
// MI455X (gfx1250) — compile-verified
//
#include <hip/hip_runtime.h>

#define IN_DIM 64
#define HID    128
#define OUT_DIM 9

typedef __attribute__((ext_vector_type(2))) float v2f;
typedef __attribute__((ext_vector_type(8))) float v8f;

// ---------------------------------------------------------------------------
// Degree: one f32 atomic per edge (non-returning global_atomic_add_f32).
// ---------------------------------------------------------------------------
__global__ void k_degree(const int* __restrict__ dst, float* __restrict__ deg, int nE) {
  int e = blockIdx.x * blockDim.x + threadIdx.x;
  if (e < nE) unsafeAtomicAdd(&deg[dst[e]], 1.0f);
}

__global__ void k_invdeg(float* __restrict__ deg, int n) {
  int i = blockIdx.x * blockDim.x + threadIdx.x;
  if (i < n) deg[i] = 1.0f / fmaxf(deg[i], 1.0f);
}

// ---------------------------------------------------------------------------
// Edge scatter: thread = (edge, float4-chunk). float4 gather (global_load_b128)
// + 4 non-returning f32 atomics into the L2-resident sum buffer.
// ---------------------------------------------------------------------------
template<int D4>  // number of float4 chunks per feature row
__global__ void k_scatter(const float* __restrict__ X,
                          const int* __restrict__ src,
                          const int* __restrict__ dst,
                          float* __restrict__ Sum, int nE) {
  long t = (long)blockIdx.x * blockDim.x + threadIdx.x;
  int e = (int)(t / D4);
  int c = (int)(t % D4);
  if (e >= nE) return;
  int s = src[e];
  int d = dst[e];
  const float4* xr = (const float4*)(X + (long)s * (D4 * 4));
  float4 v = xr[c];
  float* o = Sum + (long)d * (D4 * 4) + c * 4;
  unsafeAtomicAdd(o + 0, v.x);
  unsafeAtomicAdd(o + 1, v.y);
  unsafeAtomicAdd(o + 2, v.z);
  unsafeAtomicAdd(o + 3, v.w);
}

// ---------------------------------------------------------------------------
// Fused SAGE layer GEMM:
//   Y = relu( (Sum .* invdeg) @ Wl  +  X @ Wr  +  bias )
// Block = 256 threads = 8 waves; each wave computes a 16x16 tile of Y with
// V_WMMA_F32_16X16X4_F32.  Weight 16-col slices staged in LDS.
// WMMA f32 lane layout (ISA 7.12.2):
//   A 16x4 : lanes 0-15 rows, VGPR{0,1}=K{0,1}; lanes 16-31 same rows, K{2,3}
//   B 4x16 : lanes 0-15 cols, VGPR{0,1}=K{0,1}; lanes 16-31 same cols, K{2,3}
//   C/D    : VGPR i -> row i (lanes 0-15) / row i+8 (lanes 16-31), col = lane%16
// ---------------------------------------------------------------------------
template<int K, bool RELU>
__global__ void k_sage_gemm(const float* __restrict__ Sum,
                            const float* __restrict__ X,
                            const float* __restrict__ invdeg,
                            const float* __restrict__ Wl,
                            const float* __restrict__ Wr,
                            const float* __restrict__ bias,
                            float* __restrict__ Y,
                            int N, int DOUT) {
  __shared__ float Wls[K * 16];
  __shared__ float Wrs[K * 16];
  const int c0 = blockIdx.y * 16;

  for (int i = threadIdx.x; i < K * 16; i += blockDim.x) {
    int k = i >> 4, j = i & 15;
    Wls[i] = Wl[k * DOUT + c0 + j];
    Wrs[i] = Wr[k * DOUT + c0 + j];
  }
  __syncthreads();

  const int wave = threadIdx.x >> 5;
  const int lane = threadIdx.x & 31;
  const int hf   = lane >> 4;    // 0: K pair {0,1}; 1: K pair {2,3}
  const int lr   = lane & 15;
  const int r0   = blockIdx.x * 128 + wave * 16;
  if (r0 >= N) return;           // wave-uniform: EXEC stays all-1s for WMMA

  const int   row = r0 + lr;
  const float id  = invdeg[row];
  const float* As = Sum + (long)row * K;
  const float* Ax = X   + (long)row * K;

  v8f acc = {};
#pragma unroll 8
  for (int k = 0; k < K; k += 4) {
    const int ka = k + 2 * hf;
    v2f a, b;
    a.x = As[ka]     * id;
    a.y = As[ka + 1] * id;
    b.x = Wls[ka * 16 + lr];
    b.y = Wls[(ka + 1) * 16 + lr];
    acc = __builtin_amdgcn_wmma_f32_16x16x4_f32(false, a, false, b,
                                                (short)0, acc, false, false);
  }
#pragma unroll 8
  for (int k = 0; k < K; k += 4) {
    const int ka = k + 2 * hf;
    v2f a, b;
    a.x = Ax[ka];
    a.y = Ax[ka + 1];
    b.x = Wrs[ka * 16 + lr];
    b.y = Wrs[(ka + 1) * 16 + lr];
    acc = __builtin_amdgcn_wmma_f32_16x16x4_f32(false, a, false, b,
                                                (short)0, acc, false, false);
  }

  const float bs = bias[c0 + lr];
#pragma unroll
  for (int i = 0; i < 8; ++i) {
    float o = acc[i] + bs;
    if (RELU) o = fmaxf(o, 0.0f);
    Y[(long)(r0 + hf * 8 + i) * DOUT + c0 + lr] = o;
  }
}

// ---------------------------------------------------------------------------
// Final FC: Y[N,9] = H[N,256] @ Wfc + bfc, OUT padded to one 16-col WMMA tile.
// ---------------------------------------------------------------------------
__global__ void k_fc(const float* __restrict__ H,
                     const float* __restrict__ W,
                     const float* __restrict__ bias,
                     float* __restrict__ Y, int N) {
  constexpr int K = 2 * HID;
  __shared__ float Ws[K * 16];
  for (int i = threadIdx.x; i < K * 16; i += blockDim.x) {
    int k = i >> 4, j = i & 15;
    Ws[i] = (j < OUT_DIM) ? W[k * OUT_DIM + j] : 0.0f;
  }
  __syncthreads();

  const int wave = threadIdx.x >> 5;
  const int lane = threadIdx.x & 31;
  const int hf   = lane >> 4;
  const int lr   = lane & 15;
  const int r0   = blockIdx.x * 128 + wave * 16;
  if (r0 >= N) return;

  const float* Ah = H + (long)(r0 + lr) * K;
  v8f acc = {};
#pragma unroll 8
  for (int k = 0; k < K; k += 4) {
    const int ka = k + 2 * hf;
    v2f a, b;
    a.x = Ah[ka];
    a.y = Ah[ka + 1];
    b.x = Ws[ka * 16 + lr];
    b.y = Ws[(ka + 1) * 16 + lr];
    acc = __builtin_amdgcn_wmma_f32_16x16x4_f32(false, a, false, b,
                                                (short)0, acc, false, false);
  }

  if (lr < OUT_DIM) {
    const float bs = bias[lr];
#pragma unroll
    for (int i = 0; i < 8; ++i)
      Y[(long)(r0 + hf * 8 + i) * OUT_DIM + lr] = acc[i] + bs;
  }
}

// ---------------------------------------------------------------------------
extern "C" void kernel_launch(void* const* d_in, const int* in_sizes, int n_in,
                              void* d_out, int out_size, void* d_ws, size_t ws_size,
                              hipStream_t stream) {
  const float* x   = (const float*)d_in[0];
  const int*   ei  = (const int*)  d_in[1];
  const float* Wl1 = (const float*)d_in[2];
  const float* bl1 = (const float*)d_in[3];
  const float* Wr1 = (const float*)d_in[4];
  const float* Wl2 = (const float*)d_in[5];
  const float* bl2 = (const float*)d_in[6];
  const float* Wr2 = (const float*)d_in[7];
  const float* Wl3 = (const float*)d_in[8];
  const float* bl3 = (const float*)d_in[9];
  const float* Wr3 = (const float*)d_in[10];
  const float* Wfc = (const float*)d_in[11];
  const float* bfc = (const float*)d_in[12];
  float* out = (float*)d_out;

  const int N = in_sizes[0] / IN_DIM;   // 100000
  const int E = in_sizes[1] / 2;        // 1600000
  const int* src = ei;
  const int* dst = ei + E;

  // Workspace carve-out (all 256B aligned): invdeg | sum | h1 | h2 | h3
  char*  ws     = (char*)d_ws;
  size_t off    = 0;
  float* invdeg = (float*)(ws + off); off += ((size_t)N * 4 + 255) & ~(size_t)255;
  float* sum    = (float*)(ws + off); off += (size_t)N * HID * 4;       // 51.2 MB (L2-resident)
  float* h1     = (float*)(ws + off); off += (size_t)N * HID * 4;
  float* h2     = (float*)(ws + off); off += (size_t)N * HID * 4;
  float* h3     = (float*)(ws + off); off += (size_t)N * 2 * HID * 4;
  (void)ws_size; (void)n_in; (void)out_size;

  const int B = 256;
  dim3 gRow((N + 127) / 128);

  // Degrees -> 1/max(deg,1)
  hipMemsetAsync(invdeg, 0, (size_t)N * 4, stream);
  k_degree<<<(E + B - 1) / B, B, 0, stream>>>(dst, invdeg, E);
  k_invdeg<<<(N + B - 1) / B, B, 0, stream>>>(invdeg, N);

  // Layer 1: 64 -> 128
  hipMemsetAsync(sum, 0, (size_t)N * IN_DIM * 4, stream);
  {
    long tot = (long)E * (IN_DIM / 4);
    k_scatter<IN_DIM / 4><<<(int)((tot + B - 1) / B), B, 0, stream>>>(x, src, dst, sum, E);
  }
  k_sage_gemm<IN_DIM, true><<<dim3(gRow.x, HID / 16), B, 0, stream>>>(
      sum, x, invdeg, Wl1, Wr1, bl1, h1, N, HID);

  // Layer 2: 128 -> 128
  hipMemsetAsync(sum, 0, (size_t)N * HID * 4, stream);
  {
    long tot = (long)E * (HID / 4);
    k_scatter<HID / 4><<<(int)((tot + B - 1) / B), B, 0, stream>>>(h1, src, dst, sum, E);
  }
  k_sage_gemm<HID, true><<<dim3(gRow.x, HID / 16), B, 0, stream>>>(
      sum, h1, invdeg, Wl2, Wr2, bl2, h2, N, HID);

  // Layer 3: 128 -> 256
  hipMemsetAsync(sum, 0, (size_t)N * HID * 4, stream);
  {
    long tot = (long)E * (HID / 4);
    k_scatter<HID / 4><<<(int)((tot + B - 1) / B), B, 0, stream>>>(h2, src, dst, sum, E);
  }
  k_sage_gemm<HID, true><<<dim3(gRow.x, 2 * HID / 16), B, 0, stream>>>(
      sum, h2, invdeg, Wl3, Wr3, bl3, h3, N, 2 * HID);

  // FC: 256 -> 9
  k_fc<<<gRow, B, 0, stream>>>(h3, Wfc, bfc, out, N);
}